// VectorQuantizer_40020505264472
// MI455X (gfx1250) — compile-verified
//
#include <hip/hip_runtime.h>
#include <hip/hip_bf16.h>
#include <math.h>

typedef __attribute__((ext_vector_type(2))) float v2f;
typedef __attribute__((ext_vector_type(8))) float v8f;

#define Dk 64
#define Kn 512
#define Nn (32 * 8192)            // 262144 rows
#define NTILES (Nn / 16)          // 16384 blocks
#define DT_STRIDE 516             // padded LDS row stride (516 % 8 != 0 -> no bank overlap)

// Output layout (floats), reference tuple order:
//   vq_loss(1), quantized(64*32*8192), perplexity(1),
//   encodings(N*K), distances(N*K), indices(N)
#define OFF_QUANT 1LL
#define OFF_PERP  16777217LL
#define OFF_ENC   16777218LL
#define OFF_DIST  150994946LL
#define OFF_IDX   285212674LL

__global__ void vq_init(float* ws) {
    int t = blockIdx.x * blockDim.x + threadIdx.x;
    if (t < 513) ws[t] = 0.0f;   // ws[0] = loss accum, ws[1..512] = code counts
}

__global__ __launch_bounds__(256) void vq_main(const float* __restrict__ in,
                                               const float* __restrict__ emb,
                                               float* __restrict__ out,
                                               float* __restrict__ ws) {
    __shared__ float x2s[16];
    __shared__ float distT[16 * DT_STRIDE];
    __shared__ float redMin[256];
    __shared__ int   redIdx[256];
    __shared__ int   idxs[16];
    __shared__ float lsums[256];

    const int tid  = threadIdx.x;
    const int wave = tid >> 5;
    const int lane = tid & 31;
    const int half = lane >> 4;        // 0: K={0,1} slice, 1: K={2,3} slice
    const int r16  = lane & 15;
    const long long n0 = (long long)blockIdx.x * 16;

    float* qOut   = out + OFF_QUANT;
    float* encOut = out + OFF_ENC;
    float* dOut   = out + OFF_DIST;
    float* idxOut = out + OFF_IDX;

    // ---- Load A fragments: 16 rows x 64 d-values (striped WMMA f32 A layout) ----
    // A tile for K-step s covers d = 4s..4s+3; lanes 0-15 hold {4s,4s+1}, lanes 16-31 {4s+2,4s+3}.
    v2f a[16];
    float x2p = 0.0f;
#pragma unroll
    for (int s = 0; s < 16; ++s) {
        int d0 = 4 * s + 2 * half;
        float ax = in[(long long)d0 * Nn + n0 + r16];        // coalesced: contiguous in n
        float ay = in[(long long)(d0 + 1) * Nn + n0 + r16];
        a[s].x = ax; a[s].y = ay;
        x2p += ax * ax + ay * ay;
    }
    // complete ||x||^2 per row: other half of d's sits in lane^16
    float x2f = x2p + __shfl_xor(x2p, 16, 32);
    if (wave == 0 && half == 0) x2s[r16] = x2f;
    __syncthreads();

    // ---- 4 column tiles of 16 codebook entries per wave (wave w -> cols [64w, 64w+64)) ----
    for (int ct = 0; ct < 4; ++ct) {
        int k0 = wave * 64 + ct * 16;
        v8f acc = {};
        float e2p = 0.0f;
        const float* ebase = emb + (long long)(k0 + r16) * Dk + 2 * half;
#pragma unroll
        for (int s = 0; s < 16; ++s) {
            float bx = ebase[4 * s];
            float by = ebase[4 * s + 1];
            e2p += bx * bx + by * by;
            v2f b; b.x = bx; b.y = by;
            // D = A(16x4,f32) * B(4x16,f32) + C ; fp32 exact accumulation
            acc = __builtin_amdgcn_wmma_f32_16x16x4_f32(
                false, a[s], false, b, (short)0, acc, false, false);
        }
        float e2 = e2p + __shfl_xor(e2p, 16, 32);   // ||e||^2 for column k0+r16
#pragma unroll
        for (int v = 0; v < 8; ++v) {
            int m = v + 8 * half;                    // row within tile for this VGPR
            float dist = x2s[m] + e2 - 2.0f * acc[v];
            dOut[(n0 + m) * Kn + k0 + r16] = dist;
            distT[m * DT_STRIDE + k0 + r16] = dist;
        }
    }
    __syncthreads();

    // ---- per-row argmin over 512 distances (16 threads/row, first-min tie break) ----
    {
        int r = tid >> 4, p = tid & 15;
        float mv = 3.4e38f; int mi = 0;
        for (int j = 0; j < 32; ++j) {
            int c = p * 32 + j;
            float v = distT[r * DT_STRIDE + c];
            if (v < mv) { mv = v; mi = c; }
        }
        redMin[tid] = mv; redIdx[tid] = mi;
        __syncthreads();
        if (p == 0) {
            float bm = redMin[r * 16]; int bi = redIdx[r * 16];
            for (int q = 1; q < 16; ++q) {
                float v = redMin[r * 16 + q];
                int  i2 = redIdx[r * 16 + q];
                if (v < bm || (v == bm && i2 < bi)) { bm = v; bi = i2; }
            }
            idxs[r] = bi;
            idxOut[n0 + r] = (float)bi;
            atomicAdd(ws + 1 + bi, 1.0f);            // histogram for perplexity
        }
    }
    __syncthreads();

    // ---- one-hot encodings: each thread writes 32 contiguous floats of its row ----
    {
        int r = tid >> 4, p = tid & 15;
        int best = idxs[r];
        float2* ep = (float2*)(encOut + (n0 + r) * Kn + p * 32);  // 8B-aligned
#pragma unroll
        for (int j2 = 0; j2 < 16; ++j2) {
            int c0 = p * 32 + j2 * 2;
            float2 v;
            v.x = (c0 == best) ? 1.0f : 0.0f;
            v.y = (c0 + 1 == best) ? 1.0f : 0.0f;
            ep[j2] = v;
        }
    }

    // ---- quantized gather (straight-through forward == quantized) + loss accum ----
    {
        int rr = tid & 15, db = tid >> 4;
        int myIdx = idxs[rr];
        float lsum = 0.0f;
#pragma unroll
        for (int dd = 0; dd < 4; ++dd) {
            int d = db + 16 * dd;
            float e = emb[(long long)myIdx * Dk + d];
            long long gi = (long long)d * Nn + n0 + rr;   // (D, N) layout, coalesced in rr
            float x = in[gi];
            qOut[gi] = e;
            float df = e - x;
            lsum += df * df;
        }
        lsums[tid] = lsum;
        __syncthreads();
        for (int st = 128; st > 0; st >>= 1) {
            if (tid < st) lsums[tid] += lsums[tid + st];
            __syncthreads();
        }
        if (tid == 0) atomicAdd(ws, lsums[0]);
    }
}

__global__ __launch_bounds__(512) void vq_fin(const float* __restrict__ ws,
                                              float* __restrict__ out) {
    __shared__ float red[512];
    int t = threadIdx.x;
    float p = ws[1 + t] * (1.0f / (float)Nn);
    red[t] = p * logf(p + 1e-10f);
    __syncthreads();
    for (int st = 256; st > 0; st >>= 1) {
        if (t < st) red[t] += red[t + st];
        __syncthreads();
    }
    if (t == 0) {
        out[OFF_PERP] = expf(-red[0]);
        // q_latent_loss + 0.25 * e_latent_loss, both equal mean((q - x)^2)
        out[0] = 1.25f * ws[0] / ((float)Nn * 64.0f);
    }
}

extern "C" void kernel_launch(void* const* d_in, const int* in_sizes, int n_in,
                              void* d_out, int out_size, void* d_ws, size_t ws_size,
                              hipStream_t stream) {
    const float* inputs = (const float*)d_in[0];   // (64, 32, 8192) fp32
    const float* emb    = (const float*)d_in[1];   // (512, 64) fp32
    float* out = (float*)d_out;
    float* ws  = (float*)d_ws;

    vq_init<<<3, 256, 0, stream>>>(ws);
    vq_main<<<NTILES, 256, 0, stream>>>(inputs, emb, out, ws);
    vq_fin<<<1, 512, 0, stream>>>(ws, out);
}